// ResidualGraphBlock_12326556140089
// MI455X (gfx1250) — compile-verified
//
#include <hip/hip_runtime.h>
#include <hip/hip_bf16.h>
#include <math.h>

// ---------------------------------------------------------------------------
// ResidualGraphBlock for MI455X (gfx1250, wave32, WMMA).
// Dense math runs through V_WMMA_F32_16X16X4_F32 (full fp32 precision -- the
// workload is L2-resident and bandwidth bound, so fp32 matrix ops are free).
// Weights are staged once per workgroup into LDS in a k-pair-transposed
// layout so every B-fragment fetch is one ds_load_b64. D = 128 fixed.
// ---------------------------------------------------------------------------

typedef __attribute__((ext_vector_type(2))) float v2f;
typedef __attribute__((ext_vector_type(8))) float v8f;

#define D 128

// ---------------------------------------------------------------------------
// Generic fused GEMM:  out = act( A1@W1 [+ A2@W2] + bias ) [+ addSrc]
// One wave computes a 16x128 slab (8 WMMA tiles). Block = 8 waves = 128 rows.
// A-fragment (f32 16x4): lanes 0-15 -> K={k,k+1}, lanes 16-31 -> K={k+2,k+3}.
// B-fragment mirrors A with N in place of M; served from LDS as float2.
// C/D layout: vgpr r, lanes 0-15 -> M=r, lanes 16-31 -> M=r+8; N = lane&15.
// ---------------------------------------------------------------------------
__launch_bounds__(256)
__global__ void gemm128_wmma(const float* __restrict__ A1,
                             const float* __restrict__ W1,
                             const float* __restrict__ A2,   // may be null
                             const float* __restrict__ W2,   // may be null
                             const float* __restrict__ bias,
                             const float* __restrict__ addSrc, // may be null
                             float* __restrict__ out,
                             int nrows, int applyElu)
{
    // 64 KB LDS: Wl[p*128 + c] = ( W[2p][c], W[2p+1][c] ), p = 0..63.
    __shared__ v2f Wl[64 * D];

    const int lane = threadIdx.x & 31;
    const int wave = threadIdx.x >> 5;
    const int hi   = lane >> 4;        // which half-wave
    const int l    = lane & 15;
    const int rowBase = blockIdx.x * 128 + wave * 16;

    int aRow = rowBase + l;
    if (aRow >= nrows) aRow = nrows - 1;   // clamp so EXEC stays all-ones

    v8f acc[8] = {};                       // 8 tiles of 16x16 f32

    const int nPass = (A2 != nullptr) ? 2 : 1;
    for (int pass = 0; pass < nPass; ++pass) {
        const float* __restrict__ A = (pass == 0) ? A1 : A2;
        const float* __restrict__ W = (pass == 0) ? W1 : W2;

        // ---- Stage weights into LDS (coalesced reads, ds_store_b64) ----
        __syncthreads();                   // protect Wl readers of prev pass
        for (int i = threadIdx.x; i < 64 * D; i += 256) {
            const int p = i >> 7;          // k-pair index
            const int c = i & (D - 1);     // column
            v2f t;
            t.x = W[(size_t)(2 * p)     * D + c];
            t.y = W[(size_t)(2 * p + 1) * D + c];
            Wl[i] = t;
        }
        __syncthreads();

        // ---- MAC loop: A from global (b64 stream), B from LDS ----
        const float* __restrict__ arow = A + (size_t)aRow * D;
        for (int k = 0; k < D; k += 4) {
            const int ka = k + 2 * hi;     // even
            const int kp = ka >> 1;        // pair index
            v2f afrag;
            afrag.x = arow[ka];
            afrag.y = arow[ka + 1];
#pragma unroll
            for (int c = 0; c < 8; ++c) {
                const v2f bfrag = Wl[kp * D + c * 16 + l];   // ds_load_b64
                acc[c] = __builtin_amdgcn_wmma_f32_16x16x4_f32(
                    /*neg_a=*/false, afrag,
                    /*neg_b=*/false, bfrag,
                    /*c_mod=*/(short)0, acc[c],
                    /*reuse_a=*/false, /*reuse_b=*/false);
            }
        }
    }

    // Epilogue: bias, optional ELU, optional residual add, guarded store.
#pragma unroll
    for (int c = 0; c < 8; ++c) {
        const int col = c * 16 + l;
        const float bv = bias[col];
#pragma unroll
        for (int r = 0; r < 8; ++r) {
            const int row = rowBase + hi * 8 + r;
            if (row < nrows) {
                float v = acc[c][r] + bv;
                if (applyElu) v = (v > 0.0f) ? v : (__expf(v) - 1.0f);
                if (addSrc)   v += addSrc[(size_t)row * D + col];
                out[(size_t)row * D + col] = v;
            }
        }
    }
}

// ---------------------------------------------------------------------------
// Graph aggregation helpers
// ---------------------------------------------------------------------------
__global__ void zero_kernel(float* __restrict__ p, long long n)
{
    long long i = (long long)blockIdx.x * blockDim.x + threadIdx.x;
    if (i < n) p[i] = 0.0f;
}

__global__ void degree_kernel(const int* __restrict__ dst,
                              float* __restrict__ deg, int E)
{
    int e = blockIdx.x * blockDim.x + threadIdx.x;
    if (e < E) atomicAdd(&deg[dst[e]], 1.0f);
}

// 32 threads per edge, 4 floats per thread (float4 gather, 4 fp32 atomics).
__global__ void scatter_kernel(const float* __restrict__ feat,
                               const int* __restrict__ src,
                               const int* __restrict__ dst,
                               float* __restrict__ agg, int E)
{
    long long t = (long long)blockIdx.x * blockDim.x + threadIdx.x;
    int e    = (int)(t >> 5);
    int part = (int)(t & 31);
    if (e < E) {
        const float4 v = *(const float4*)(feat + (size_t)src[e] * D + part * 4);
        float* o = agg + (size_t)dst[e] * D + part * 4;
        atomicAdd(o + 0, v.x);
        atomicAdd(o + 1, v.y);
        atomicAdd(o + 2, v.z);
        atomicAdd(o + 3, v.w);
    }
}

__global__ void normalize_kernel(float* __restrict__ agg,
                                 const float* __restrict__ deg, int N)
{
    long long t = (long long)blockIdx.x * blockDim.x + threadIdx.x;
    int i    = (int)(t >> 5);
    int part = (int)(t & 31);
    if (i < N) {
        float d = deg[i];
        float inv = 1.0f / ((d > 1.0f) ? d : 1.0f);
        float4* p = (float4*)(agg + (size_t)i * D + part * 4);
        float4 v = *p;
        v.x *= inv; v.y *= inv; v.z *= inv; v.w *= inv;
        *p = v;
    }
}

// ---------------------------------------------------------------------------
// Host-side orchestration (graph-capture safe: kernels only, all on stream)
// ---------------------------------------------------------------------------
extern "C" void kernel_launch(void* const* d_in, const int* in_sizes, int n_in,
                              void* d_out, int out_size, void* d_ws, size_t ws_size,
                              hipStream_t stream)
{
    const float* h        = (const float*)d_in[0];
    const int*   src      = (const int*)  d_in[1];
    const int*   dst      = (const int*)  d_in[2];
    const float* W_skip1  = (const float*)d_in[3];
    const float* b_skip1  = (const float*)d_in[4];
    const float* W_skip2  = (const float*)d_in[5];
    const float* b_skip2  = (const float*)d_in[6];
    const float* W_self1  = (const float*)d_in[7];
    const float* W_neigh1 = (const float*)d_in[8];
    const float* b_conv1  = (const float*)d_in[9];
    const float* W_si1    = (const float*)d_in[10];
    const float* b_si1    = (const float*)d_in[11];
    const float* W_si2    = (const float*)d_in[12];
    const float* b_si2    = (const float*)d_in[13];
    const float* W_self2  = (const float*)d_in[14];
    const float* W_neigh2 = (const float*)d_in[15];
    const float* b_conv2  = (const float*)d_in[16];

    const int N = in_sizes[0] / D;
    const int E = in_sizes[1];

    // Workspace carve: deg[N] + three N x 128 fp32 buffers.
    char* w = (char*)d_ws;
    const size_t degBytes = ((size_t)N * sizeof(float) + 255) & ~(size_t)255;
    float* deg = (float*)w;
    float* B1  = (float*)(w + degBytes);          // rotating buffer
    float* B2  = B1 + (size_t)N * D;              // h_skip (persists to end)
    float* B3  = B2 + (size_t)N * D;              // rotating buffer
    float* outp = (float*)d_out;

    const dim3 blk(256);
    const int gemmGrid = (N + 127) / 128;
    const long long featN = (long long)N * D;
    const int zeroGridF = (int)((featN + 255) / 256);
    const int zeroGridD = (N + 255) / 256;
    const int edgeGrid  = (E + 255) / 256;
    const int edgeFeatGrid = (int)(((long long)E * 32 + 255) / 256);
    const int nodeFeatGrid = (int)(((long long)N * 32 + 255) / 256);

    // --- Skip MLP: h_skip = ELU(h@W_skip1+b) @ W_skip2 + b  -> B2 ---
    gemm128_wmma<<<gemmGrid, blk, 0, stream>>>(h, W_skip1, nullptr, nullptr,
                                               b_skip1, nullptr, B1, N, 1);
    gemm128_wmma<<<gemmGrid, blk, 0, stream>>>(B1, W_skip2, nullptr, nullptr,
                                               b_skip2, nullptr, B2, N, 0);

    // --- Aggregation 1: agg1 = mean_{in-neighbors} h  -> B1 ---
    zero_kernel<<<zeroGridD, blk, 0, stream>>>(deg, N);
    zero_kernel<<<zeroGridF, blk, 0, stream>>>(B1, featN);
    degree_kernel<<<edgeGrid, blk, 0, stream>>>(dst, deg, E);
    scatter_kernel<<<edgeFeatGrid, blk, 0, stream>>>(h, src, dst, B1, E);
    normalize_kernel<<<nodeFeatGrid, blk, 0, stream>>>(B1, deg, N);

    // --- SAGEConv1: t1 = ELU(h@W_self1 + agg1@W_neigh1 + b_conv1) -> B3 ---
    gemm128_wmma<<<gemmGrid, blk, 0, stream>>>(h, W_self1, B1, W_neigh1,
                                               b_conv1, nullptr, B3, N, 1);

    // --- Self-interaction MLP: x = ELU(ELU(t1@W_si1+b)@W_si2+b) ---
    gemm128_wmma<<<gemmGrid, blk, 0, stream>>>(B3, W_si1, nullptr, nullptr,
                                               b_si1, nullptr, B1, N, 1);
    gemm128_wmma<<<gemmGrid, blk, 0, stream>>>(B1, W_si2, nullptr, nullptr,
                                               b_si2, nullptr, B3, N, 1);   // x -> B3

    // --- Aggregation 2: agg2 = mean_{in-neighbors} x -> B1 (deg unchanged) ---
    zero_kernel<<<zeroGridF, blk, 0, stream>>>(B1, featN);
    scatter_kernel<<<edgeFeatGrid, blk, 0, stream>>>(B3, src, dst, B1, E);
    normalize_kernel<<<nodeFeatGrid, blk, 0, stream>>>(B1, deg, N);

    // --- SAGEConv2 + residual: out = x@W_self2 + agg2@W_neigh2 + b + h_skip ---
    gemm128_wmma<<<gemmGrid, blk, 0, stream>>>(B3, W_self2, B1, W_neigh2,
                                               b_conv2, B2, outp, N, 0);
}